// ModelNew_4647154615082
// MI455X (gfx1250) — compile-verified
//
#include <hip/hip_runtime.h>
#include <hip/hip_bf16.h>

// Problem constants (from reference): H=2048, I=4096, E=8, TOPK=2, B=2, S=2048
#define HDIM   2048
#define IDIM   4096
#define NEXP   8
#define TOPKN  2
#define NTOK   4096                 // B*S
#define NPAIR  (NTOK * TOPKN)       // 8192 (token,slot) pairs
#define CAP    NPAIR                // worst-case tokens per expert
#define TILE_M 16                   // tokens per workgroup tile
#define MAX_TILES (NPAIR / TILE_M)  // 512 worst-case tiles per expert
#define ICHUNK 128                  // I-dimension chunk (16 rows per wave, 8 waves)
#define XS     2052                 // padded LDS stride for X rows (bank-conflict-free, 16B aligned)
#define IS2    132                  // padded LDS stride for inter^T columns (col-major [16 tok][IS2])

typedef __attribute__((ext_vector_type(2))) float v2f;
typedef __attribute__((ext_vector_type(8))) float v8f;

__device__ __forceinline__ v8f wmma_f32(v2f a, v2f b, v8f c) {
  // D[16x16] += A[16x4] * B[4x16], fp32, wave32
  return __builtin_amdgcn_wmma_f32_16x16x4_f32(false, a, false, b, (short)0, c,
                                               false, false);
}

__global__ void zero_init(float* __restrict__ out, long n, int* __restrict__ cnt) {
  long i = (long)blockIdx.x * blockDim.x + threadIdx.x;
  long stride = (long)gridDim.x * blockDim.x;
  float4 z = make_float4(0.f, 0.f, 0.f, 0.f);
  for (long k = i; k < n / 4; k += stride) ((float4*)out)[k] = z;
  if (i < NEXP) cnt[i] = 0;
}

__global__ void route_kernel(const int* __restrict__ idx, const float* __restrict__ w,
                             int* __restrict__ cnt, int* __restrict__ toks,
                             float* __restrict__ tws) {
  int i = blockIdx.x * blockDim.x + threadIdx.x;
  if (i >= NPAIR) return;
  int e = idx[i];
  int pos = atomicAdd(&cnt[e], 1);
  toks[e * CAP + pos] = i / TOPKN;   // token id
  tws[e * CAP + pos]  = w[i];        // slot weight
}

__global__ __launch_bounds__(256)
void moe_ffn_kernel(const float* __restrict__ x,
                    const float* __restrict__ gate,
                    const float* __restrict__ up,
                    const float* __restrict__ down,
                    const int* __restrict__ cnt,
                    const int* __restrict__ toks,
                    const float* __restrict__ tws,
                    float* __restrict__ out) {
  extern __shared__ float smem[];
  float* xs   = smem;                    // [16][XS]   staged X tile
  float* is2  = xs + TILE_M * XS;        // [16][IS2]  inter^T, column-major (col=token)
  int*   s_tok = (int*)(is2 + TILE_M * IS2);
  float* s_w   = (float*)(s_tok + TILE_M);

  const int e    = blockIdx.x / MAX_TILES;
  const int tile = blockIdx.x % MAX_TILES;
  const int count = cnt[e];
  if (tile * TILE_M >= count) return;    // block-uniform early exit

  const int tid  = threadIdx.x;
  const int lane = tid & 31;
  const int wave = tid >> 5;             // 0..7
  const int n    = lane & 15;            // WMMA N column / A row (M)
  const int kh   = lane >> 4;            // K-pair selector (0 or 1)

  if (tid < TILE_M) {
    int slot = tile * TILE_M + tid;
    if (slot < count) {
      s_tok[tid] = toks[e * CAP + slot];
      s_w[tid]   = tws[e * CAP + slot];
    } else {                             // tail padding: replicate a valid token, weight 0
      s_tok[tid] = toks[e * CAP + tile * TILE_M];
      s_w[tid]   = 0.0f;
    }
  }
  __syncthreads();

  // Stage X tile [16, 2048] into LDS with async DMA (gfx1250 ASYNCcnt path).
  // Each b128 transfer moves 16B: global (gathered rows) -> LDS, bypassing VGPRs.
  for (int q = tid; q < TILE_M * (HDIM / 4); q += 256) {
    int r = q / (HDIM / 4);
    int c = q % (HDIM / 4);
    unsigned      lds = (unsigned)(uintptr_t)(xs + r * XS + c * 4);     // LDS aperture: low 32 bits
    unsigned long ga  = (unsigned long)(uintptr_t)(x + (size_t)s_tok[r] * HDIM + c * 4);
    asm volatile("global_load_async_to_lds_b128 %0, %1, off"
                 :: "v"(lds), "v"(ga) : "memory");
  }
  asm volatile("s_wait_asynccnt 0x0" ::: "memory");
  __syncthreads();

  const int   tokcol = s_tok[n];
  const float wcol   = s_w[n];

  v8f oacc[16];                          // out^T rows [wave*256, wave*256+256) x 16 tokens
  #pragma unroll
  for (int jt = 0; jt < 16; ++jt) oacc[jt] = (v8f){};

  const float* Gbase = gate + (size_t)e * IDIM * HDIM;
  const float* Ubase = up   + (size_t)e * IDIM * HDIM;
  const float* Dbase = down + (size_t)e * HDIM * IDIM;

  for (int ic = 0; ic < IDIM; ic += ICHUNK) {
    // ---- gate/up: g^T tile = G[rows ic+wave*16 .. +16) x X^T, K over H ----
    const int irow = ic + wave * 16 + n;     // A row (M = lane&15)
    const float* __restrict__ Grow = Gbase + (size_t)irow * HDIM + 2 * kh;
    const float* __restrict__ Urow = Ubase + (size_t)irow * HDIM + 2 * kh;
    const float* __restrict__ Xrow = xs + n * XS + 2 * kh;

    v8f gacc = (v8f){};
    v8f uacc = (v8f){};
    __builtin_prefetch(Grow, 0, 0);
    __builtin_prefetch(Urow, 0, 0);
    #pragma unroll 8
    for (int h = 0; h < HDIM; h += 4) {
      v2f b  = *(const v2f*)(Xrow + h);      // B[k][n] = X[tok_n, h+k] (LDS, conflict-free)
      v2f ag = *(const v2f*)(Grow + h);      // A[m][k] = G[irow, h+k]  (row stream)
      v2f au = *(const v2f*)(Urow + h);
      gacc = wmma_f32(ag, b, gacc);
      uacc = wmma_f32(au, b, uacc);
    }

    // ---- fused SiLU(g)*u, write inter^T chunk to LDS (col-major) ----
    __syncthreads();                         // previous chunk's is2 reads done
    #pragma unroll
    for (int r = 0; r < 8; ++r) {
      float g  = gacc[r];
      // silu(g) = g * sigmoid(g); fast 1-ulp v_rcp_f32 instead of IEEE divide
      float sg = g * __builtin_amdgcn_rcpf(1.0f + __expf(-g));
      int   m  = wave * 16 + r + kh * 8;     // local I-row of this element
      is2[n * IS2 + m] = sg * uacc[r];
    }
    __syncthreads();

    // ---- down: out^T[j, t] += D[j, ic..ic+128) @ inter^T ----
    #pragma unroll
    for (int jt = 0; jt < 16; ++jt) {
      const int j = wave * 256 + jt * 16 + n;                 // A row (M = lane&15)
      const float* __restrict__ Drow = Dbase + (size_t)j * IDIM + ic + 2 * kh;
      __builtin_prefetch(Drow + ICHUNK, 0, 0);                // next chunk of this row
      v8f acc = oacc[jt];
      #pragma unroll 4
      for (int k0 = 0; k0 < ICHUNK; k0 += 4) {
        v2f b = *(const v2f*)(&is2[n * IS2 + k0 + 2 * kh]);   // B[k][n] = inter^T[k, tok_n]
        v2f a = *(const v2f*)(Drow + k0);                     // A[m][k] = D[j, ic+k]
        acc = wmma_f32(a, b, acc);
      }
      oacc[jt] = acc;
    }
  }

  // ---- weighted scatter: out[tok, j] += w[tok] * out^T[j, tok] ----
  const size_t obase = (size_t)tokcol * HDIM;
  #pragma unroll
  for (int jt = 0; jt < 16; ++jt) {
    #pragma unroll
    for (int r = 0; r < 8; ++r) {
      int j = wave * 256 + jt * 16 + r + kh * 8;
      atomicAdd(&out[obase + j], oacc[jt][r] * wcol);
    }
  }
}

extern "C" void kernel_launch(void* const* d_in, const int* in_sizes, int n_in,
                              void* d_out, int out_size, void* d_ws, size_t ws_size,
                              hipStream_t stream) {
  const float* x    = (const float*)d_in[0];
  const int*   idx  = (const int*)d_in[1];     // expert_indices
  const float* w    = (const float*)d_in[2];   // expert_weights
  const float* gate = (const float*)d_in[3];   // [E, I, H]
  const float* up   = (const float*)d_in[4];   // [E, I, H]
  const float* down = (const float*)d_in[5];   // [E, H, I]
  float* out = (float*)d_out;

  char* ws   = (char*)d_ws;
  int*   cnt  = (int*)ws;                                  // 8 counters
  int*   toks = (int*)(ws + 256);                          // E*CAP ints (256 KB)
  float* tws  = (float*)(ws + 256 + (size_t)NEXP * CAP * sizeof(int));

  zero_init<<<2048, 256, 0, stream>>>(out, (long)NTOK * HDIM, cnt);
  route_kernel<<<(NPAIR + 255) / 256, 256, 0, stream>>>(idx, w, cnt, toks, tws);

  size_t smem = (TILE_M * XS + TILE_M * IS2 + 2 * TILE_M) * sizeof(float); // ~137 KB
  moe_ffn_kernel<<<NEXP * MAX_TILES, 256, smem, stream>>>(x, gate, up, down,
                                                          cnt, toks, tws, out);
}